// VectorQuantizerEMA_14834817040644
// MI455X (gfx1250) — compile-verified
//
#include <hip/hip_runtime.h>
#include <math.h>

typedef __attribute__((ext_vector_type(2))) float v2f;
typedef __attribute__((ext_vector_type(8))) float v8f;

#define K_CODES 1024
#define DIM 64
#define N_VEC 131072      // 32*64*64 rows
#define Z_ELEMS 8388608   // 32*64*64*64
#define ROW_STRIDE 66     // LDS row pad: conflict-free, keeps b64 8B-aligned

// ws float layout: [0]=loss accum, [1..1024]=counts, [1025..2048]=e_sq

// Branchless argmin-pair update: sets (d,i) to (nd,ni) if (nd,ni) beats (d,i).
__device__ __forceinline__ void argmin_update(float& d, int& i, float nd, int ni) {
    // bitwise combine to avoid short-circuit control flow -> v_cmp + v_cndmask
    const int better = (int)(nd < d) | ((int)(nd == d) & (int)(ni < i));
    d = better ? nd : d;
    i = better ? ni : i;
}

__global__ __launch_bounds__(256) void vq_prep(const float* __restrict__ emb,
                                               float* __restrict__ ws) {
    int j = blockIdx.x * 256 + threadIdx.x;   // 0..1023
    if (j < K_CODES) {
        const float* e = emb + j * DIM;
        float s = 0.f;
        #pragma unroll
        for (int c = 0; c < DIM; ++c) s += e[c] * e[c];
        ws[1025 + j] = s;
        ws[1 + j]    = 0.f;   // zero histogram every launch
    }
    if (j == 0) ws[0] = 0.f;  // zero loss accumulator
}

__global__ __launch_bounds__(256) void vq_main(const float* __restrict__ z,
                                               const float* __restrict__ emb,
                                               float* __restrict__ ws,
                                               float* __restrict__ out) {
    __shared__ float lds_z[64 * ROW_STRIDE];
    __shared__ float pmin[64][8];
    __shared__ int   pidx[64][8];
    __shared__ int   lds_code[64];
    __shared__ float lsum[8];

    const int tid  = threadIdx.x;
    const int lane = tid & 31;
    const int wave = tid >> 5;
    const int b    = blockIdx.x >> 6;
    const int h    = blockIdx.x & 63;
    const long zbase = (long)b * 262144 + (long)h * 64;   // z[b, c, h, w] = zbase + c*4096 + w

    // ---- Phase A: transpose-load z tile into LDS as [w][c] ----
    {
        const int w  = tid & 63;
        const int cg = tid >> 6;                // 0..3
        const float* zp = z + zbase + w;
        #pragma unroll
        for (int i = 0; i < 16; ++i) {
            const int c = cg * 16 + i;
            lds_z[w * ROW_STRIDE + c] = zp[(long)c * 4096];
        }
    }
    __syncthreads();

    // ---- Phase B: scores via V_WMMA_F32_16X16X4_F32, running argmin ----
    const int nlo   = lane & 15;          // column within tile / A-row within half
    const int khalf = (lane >> 4) * 2;    // K offset: lanes 16-31 hold K+2,K+3

    float mind[4][8];
    int   mini[4][8];
    #pragma unroll
    for (int rt = 0; rt < 4; ++rt)
        #pragma unroll
        for (int r = 0; r < 8; ++r) { mind[rt][r] = 3.4e38f; mini[rt][r] = 0x7fffffff; }

    const float* wsq = ws + 1025;

    for (int j = 0; j < 8; ++j) {
        const int ct = wave + 8 * j;          // this wave's column tile
        const int e0 = ct * 16;
        const int ci = e0 + nlo;              // absolute codebook index this lane scores

        // B fragments: lane nlo holds codebook entry e0+nlo, channel pair khalf+4*ks
        v2f bfrag[16];
        const float* bbase = emb + (long)ci * DIM + khalf;
        #pragma unroll
        for (int ks = 0; ks < 16; ++ks)
            bfrag[ks] = *(const v2f*)(bbase + 4 * ks);
        const float esq = wsq[ci];

        #pragma unroll
        for (int rt = 0; rt < 4; ++rt) {
            v8f acc = {0.f, 0.f, 0.f, 0.f, 0.f, 0.f, 0.f, 0.f};
            const float* abase = &lds_z[(rt * 16 + nlo) * ROW_STRIDE + khalf];
            #pragma unroll
            for (int ks = 0; ks < 16; ++ks) {
                v2f a = *(const v2f*)(abase + 4 * ks);
                acc = __builtin_amdgcn_wmma_f32_16x16x4_f32(
                    false, a, false, bfrag[ks], (short)0, acc, false, false);
            }
            #pragma unroll
            for (int r = 0; r < 8; ++r) {
                const float s = __builtin_fmaf(-2.0f, acc[r], esq);
                argmin_update(mind[rt][r], mini[rt][r], s, ci);
            }
        }
    }

    // ---- butterfly argmin across the 16 lanes sharing each row ----
    #pragma unroll
    for (int rt = 0; rt < 4; ++rt) {
        #pragma unroll
        for (int r = 0; r < 8; ++r) {
            float d  = mind[rt][r];
            int   ix = mini[rt][r];
            #pragma unroll
            for (int off = 1; off < 16; off <<= 1) {
                const float od = __shfl_xor(d, off, 32);
                const int   oi = __shfl_xor(ix, off, 32);
                argmin_update(d, ix, od, oi);
            }
            if (nlo == 0) {   // lanes 0 and 16: rows rt*16+r and rt*16+8+r
                const int row = rt * 16 + r + (lane >> 4) * 8;
                pmin[row][wave] = d;
                pidx[row][wave] = ix;
            }
        }
    }
    __syncthreads();

    // ---- final per-row argmin over the 8 waves; codes + histogram ----
    if (tid < 64) {
        float best = pmin[tid][0];
        int   bi   = pidx[tid][0];
        #pragma unroll
        for (int v = 1; v < 8; ++v)
            argmin_update(best, bi, pmin[tid][v], pidx[tid][v]);
        lds_code[tid] = bi;
        out[Z_ELEMS + 2 + (long)blockIdx.x * 64 + tid] = (float)bi;  // codes
        atomicAdd(&ws[1 + bi], 1.0f);
    }
    __syncthreads();

    // ---- Phase C: gather quantized, write back [B,C,H,W], accumulate MSE ----
    float acc_loss = 0.f;
    {
        const int w   = tid & 63;
        const int cg  = tid >> 6;
        const int idx = lds_code[w];
        const float* e = emb + (long)idx * DIM;
        float* op = out + zbase + w;
        #pragma unroll
        for (int i = 0; i < 16; ++i) {
            const int c = cg * 16 + i;
            const float q = e[c];
            const float d = q - lds_z[w * ROW_STRIDE + c];
            acc_loss = __builtin_fmaf(d, d, acc_loss);
            op[(long)c * 4096] = q;
        }
    }
    #pragma unroll
    for (int off = 1; off < 32; off <<= 1)
        acc_loss += __shfl_xor(acc_loss, off, 32);
    if (lane == 0) lsum[wave] = acc_loss;
    __syncthreads();
    if (tid == 0) {
        float s = 0.f;
        #pragma unroll
        for (int v = 0; v < 8; ++v) s += lsum[v];
        atomicAdd(&ws[0], s);
    }
}

__global__ __launch_bounds__(256) void vq_finalize(const float* __restrict__ ws,
                                                   float* __restrict__ out) {
    __shared__ float red[8];
    const int tid = threadIdx.x;
    float local = 0.f;
    for (int j = tid; j < K_CODES; j += 256) {
        const float p = ws[1 + j] * (1.0f / (float)N_VEC);
        local += p * logf(p + 1e-5f);
    }
    #pragma unroll
    for (int off = 1; off < 32; off <<= 1)
        local += __shfl_xor(local, off, 32);
    if ((tid & 31) == 0) red[tid >> 5] = local;
    __syncthreads();
    if (tid == 0) {
        float s = 0.f;
        #pragma unroll
        for (int v = 0; v < 8; ++v) s += red[v];
        out[Z_ELEMS]     = ws[0] * (0.25f / (float)Z_ELEMS);  // vq_loss
        out[Z_ELEMS + 1] = expf(-s);                          // perplexity
    }
}

extern "C" void kernel_launch(void* const* d_in, const int* in_sizes, int n_in,
                              void* d_out, int out_size, void* d_ws, size_t ws_size,
                              hipStream_t stream) {
    (void)in_sizes; (void)n_in; (void)out_size; (void)ws_size;
    const float* z   = (const float*)d_in[0];
    const float* emb = (const float*)d_in[1];
    float* out = (float*)d_out;
    float* ws  = (float*)d_ws;

    vq_prep<<<4, 256, 0, stream>>>(emb, ws);
    vq_main<<<2048, 256, 0, stream>>>(z, emb, ws, out);
    vq_finalize<<<1, 256, 0, stream>>>(ws, out);
}